// ZengGNN_19559281066123
// MI455X (gfx1250) — compile-verified
//
#include <hip/hip_runtime.h>

#define N_NODES   50000
#define N_EDGES   800000
#define K_HOPS    4
#define N_LAYERS  3
#define NFEAT     128
#define HOP_DIM   32
#define N_CLASSES 40

typedef __attribute__((ext_vector_type(2))) float v2f;
typedef __attribute__((ext_vector_type(8))) float v8f;

// proj[hop][n][0:32] = h[n][0:128] @ W[l][hop]   (f32 WMMA 16x16x4)
__global__ __launch_bounds__(256) void gemm_proj(const float* __restrict__ h,
                                                 const float* __restrict__ Wl,   // [K_HOPS][128][32]
                                                 float* __restrict__ proj)       // [K_HOPS][N][32]
{
    __shared__ float sW[NFEAT * HOP_DIM];  // 16 KB
    const int hop = blockIdx.y;
    const float* Wk = Wl + hop * NFEAT * HOP_DIM;
    for (int i = threadIdx.x; i < NFEAT * HOP_DIM; i += 256) sW[i] = Wk[i];
    __syncthreads();

    const int wave = threadIdx.x >> 5;
    const int lane = threadIdx.x & 31;
    const int lo   = lane & 15;
    const int hi   = lane >> 4;
    const int mtile = blockIdx.x * 8 + wave;
    if (mtile >= N_NODES / 16) return;          // wave-uniform: EXEC stays all-1s for WMMA
    const int m0 = mtile * 16;

    v8f acc0 = {}; v8f acc1 = {};
    const float* Arow = h + (size_t)(m0 + lo) * NFEAT;
    #pragma unroll 4
    for (int k = 0; k < NFEAT; k += 4) {
        const int kk = k + 2 * hi;
        v2f a;  a.x  = Arow[kk];                 a.y  = Arow[kk + 1];
        v2f b0; b0.x = sW[kk * HOP_DIM + lo];    b0.y = sW[(kk + 1) * HOP_DIM + lo];
        v2f b1; b1.x = sW[kk * HOP_DIM + 16 + lo]; b1.y = sW[(kk + 1) * HOP_DIM + 16 + lo];
        acc0 = __builtin_amdgcn_wmma_f32_16x16x4_f32(false, a, false, b0, (short)0, acc0, false, false);
        acc1 = __builtin_amdgcn_wmma_f32_16x16x4_f32(false, a, false, b1, (short)0, acc1, false, false);
    }

    float* P = proj + (size_t)hop * N_NODES * HOP_DIM;
    #pragma unroll
    for (int r = 0; r < 8; ++r) {
        const int m = m0 + r + 8 * hi;
        P[(size_t)m * HOP_DIM + lo]      = acc0[r];
        P[(size_t)m * HOP_DIM + 16 + lo] = acc1[r];
    }
}

// h_out[n][j] = bias_l[j]   (bias_l = b[l] flattened: [K_HOPS*HOP_DIM] == [128])
__global__ __launch_bounds__(256) void bias_init(const float* __restrict__ bl,
                                                 float* __restrict__ h_out)
{
    const int i = blockIdx.x * 256 + threadIdx.x;
    if (i < N_NODES * NFEAT) h_out[i] = bl[i & (NFEAT - 1)];
}

// h_out[dst][hop*32 + c] += val * proj[hop][src][c]   (one wave per edge, one lane per channel)
__global__ __launch_bounds__(256) void spmm_scatter(const float* __restrict__ proj,
                                                    const float* __restrict__ edge_val,
                                                    const int* __restrict__ edge_src,
                                                    const int* __restrict__ edge_dst,
                                                    float* __restrict__ h_out)
{
    const int hop = blockIdx.y;
    const int t   = blockIdx.x * 256 + threadIdx.x;
    const int e   = t >> 5;
    const int c   = t & 31;
    if (e >= N_EDGES) return;
    const int eh  = hop * N_EDGES + e;
    const int src = edge_src[eh];
    const int dst = edge_dst[eh];
    const float v = edge_val[eh];
    const float contrib = v * proj[((size_t)hop * N_NODES + src) * HOP_DIM + c];
    unsafeAtomicAdd(&h_out[(size_t)dst * NFEAT + hop * HOP_DIM + c], contrib);
}

// out[n][0:40] = h[n][0:128] @ clf_W + clf_b   (N padded 40 -> 48 in LDS)
__global__ __launch_bounds__(256) void gemm_clf(const float* __restrict__ h,
                                                const float* __restrict__ clfW,   // [128][40]
                                                const float* __restrict__ clfb,   // [40]
                                                float* __restrict__ out)          // [N][40]
{
    __shared__ float sW[NFEAT * 48];  // 24 KB, zero-padded columns 40..47
    for (int i = threadIdx.x; i < NFEAT * 48; i += 256) {
        const int row = i / 48, col = i % 48;
        sW[i] = (col < N_CLASSES) ? clfW[row * N_CLASSES + col] : 0.0f;
    }
    __syncthreads();

    const int wave = threadIdx.x >> 5;
    const int lane = threadIdx.x & 31;
    const int lo   = lane & 15;
    const int hi   = lane >> 4;
    const int mtile = blockIdx.x * 8 + wave;
    if (mtile >= N_NODES / 16) return;
    const int m0 = mtile * 16;

    v8f acc0 = {}; v8f acc1 = {}; v8f acc2 = {};
    const float* Arow = h + (size_t)(m0 + lo) * NFEAT;
    #pragma unroll 4
    for (int k = 0; k < NFEAT; k += 4) {
        const int kk = k + 2 * hi;
        v2f a;  a.x  = Arow[kk];                    a.y  = Arow[kk + 1];
        v2f b0; b0.x = sW[kk * 48 + lo];            b0.y = sW[(kk + 1) * 48 + lo];
        v2f b1; b1.x = sW[kk * 48 + 16 + lo];       b1.y = sW[(kk + 1) * 48 + 16 + lo];
        v2f b2; b2.x = sW[kk * 48 + 32 + lo];       b2.y = sW[(kk + 1) * 48 + 32 + lo];
        acc0 = __builtin_amdgcn_wmma_f32_16x16x4_f32(false, a, false, b0, (short)0, acc0, false, false);
        acc1 = __builtin_amdgcn_wmma_f32_16x16x4_f32(false, a, false, b1, (short)0, acc1, false, false);
        acc2 = __builtin_amdgcn_wmma_f32_16x16x4_f32(false, a, false, b2, (short)0, acc2, false, false);
    }

    #pragma unroll
    for (int r = 0; r < 8; ++r) {
        const int m = m0 + r + 8 * hi;
        out[(size_t)m * N_CLASSES + lo]      = acc0[r] + clfb[lo];
        out[(size_t)m * N_CLASSES + 16 + lo] = acc1[r] + clfb[16 + lo];
        const int col = 32 + lo;
        if (col < N_CLASSES)
            out[(size_t)m * N_CLASSES + col] = acc2[r] + clfb[col];
    }
}

extern "C" void kernel_launch(void* const* d_in, const int* in_sizes, int n_in,
                              void* d_out, int out_size, void* d_ws, size_t ws_size,
                              hipStream_t stream) {
    const float* x        = (const float*)d_in[0];   // [N][128]
    const float* edge_val = (const float*)d_in[1];   // [4][E]
    const float* W        = (const float*)d_in[2];   // [3][4][128][32]
    const float* b        = (const float*)d_in[3];   // [3][4][32]
    const float* clf_W    = (const float*)d_in[4];   // [128][40]
    const float* clf_b    = (const float*)d_in[5];   // [40]
    const int*   edge_src = (const int*)d_in[6];     // [4][E]
    const int*   edge_dst = (const int*)d_in[7];     // [4][E]
    float* out = (float*)d_out;

    const size_t HN = (size_t)N_NODES * NFEAT;       // 6.4M floats
    float* bufA = (float*)d_ws;
    float* bufB = bufA + HN;
    float* proj = bufB + HN;                          // [4][N][32] = 6.4M floats

    const int MTILES = N_NODES / 16;                  // 3125 (exact)
    const dim3 gGemm((MTILES + 7) / 8, K_HOPS);
    const dim3 gBias((N_NODES * NFEAT + 255) / 256);
    const dim3 gScat((N_EDGES * 32) / 256, K_HOPS);

    for (int l = 0; l < N_LAYERS; ++l) {
        const float* h_in = (l == 0) ? x    : (l == 1 ? bufA : bufB);
        float*      h_out = (l == 0) ? bufA : (l == 1 ? bufB : bufA);

        gemm_proj<<<gGemm, 256, 0, stream>>>(
            h_in, W + (size_t)l * K_HOPS * NFEAT * HOP_DIM, proj);
        bias_init<<<gBias, 256, 0, stream>>>(
            b + (size_t)l * K_HOPS * HOP_DIM, h_out);
        spmm_scatter<<<gScat, 256, 0, stream>>>(
            proj, edge_val, edge_src, edge_dst, h_out);
    }

    gemm_clf<<<dim3((MTILES + 7) / 8), 256, 0, stream>>>(bufA, clf_W, clf_b, out);
}